// RPNLoss_40226663695030
// MI455X (gfx1250) — compile-verified
//
#include <hip/hip_runtime.h>

// RPN loss (class CE + probabilistic regression) for gfx1250 (MI455X).
// Memory-bound gather-reduce: ~64 B + ~50 FLOP per valid anchor -> f32,
// 128-bit gathers, deterministic two-level tree reduction (no atomics).
// Cross-lane sums use v_wmma_f32_16x16x4_f32 (A = ones -> column sums of B),
// which is exact and layout-robust, then 4 xor-shuffles finish the 32 lanes.

typedef __attribute__((ext_vector_type(2))) float v2f;
typedef __attribute__((ext_vector_type(8))) float v8f;

#define RPN_EPS 0.001f
#define BLK 256
#define WAVES_PER_BLK (BLK / 32)

// Sum x across all 32 lanes of the wave; every lane returns the total.
// EXEC must be all ones (callers guarantee full-EXEC predication).
__device__ __forceinline__ float wave_sum32(float x) {
    v2f ones; ones[0] = 1.0f; ones[1] = 1.0f;
    v2f b;    b[0] = x;       b[1] = 0.0f;
    v8f c = {};
    // D[m][n] = sum_k A[m][k]*B[k][n] = sum_k B[k][n]  (A all ones)
    // => sum over columns of row 0 == sum of all 64 B elements == sum_lanes x.
    v8f d = __builtin_amdgcn_wmma_f32_16x16x4_f32(
        /*neg_a=*/false, ones, /*neg_b=*/false, b,
        /*c_mod=*/(short)0, c, /*reuse_a=*/false, /*reuse_b=*/false);
    float s = d[0]; // lane n (0-15): D[0][n]; lane n (16-31): D[8][n-16] (rows identical)
    s += __shfl_xor(s, 1, 32);
    s += __shfl_xor(s, 2, 32);
    s += __shfl_xor(s, 4, 32);
    s += __shfl_xor(s, 8, 32);
    return s;
}

// Deterministic block-wide reduction of 4 accumulators.
// Returns true on thread 0 with r[0..3] filled.
__device__ __forceinline__ bool block_reduce4(float a0, float a1, float a2, float a3,
                                              float r[4]) {
    __shared__ float sm[WAVES_PER_BLK][4];
    float s0 = wave_sum32(a0);
    float s1 = wave_sum32(a1);
    float s2 = wave_sum32(a2);
    float s3 = wave_sum32(a3);
    const int wave = threadIdx.x >> 5;
    const int lane = threadIdx.x & 31;
    if (lane == 0) {
        sm[wave][0] = s0; sm[wave][1] = s1; sm[wave][2] = s2; sm[wave][3] = s3;
    }
    __syncthreads();
    if (threadIdx.x == 0) {
        float t0 = 0.f, t1 = 0.f, t2 = 0.f, t3 = 0.f;
#pragma unroll
        for (int w = 0; w < WAVES_PER_BLK; ++w) {  // fixed order -> deterministic
            t0 += sm[w][0]; t1 += sm[w][1]; t2 += sm[w][2]; t3 += sm[w][3];
        }
        r[0] = t0; r[1] = t1; r[2] = t2; r[3] = t3;
        return true;
    }
    return false;
}

__global__ void __launch_bounds__(BLK)
rpn_partial_kernel(const float* __restrict__ pred,
                   const float* __restrict__ unc,
                   const float* __restrict__ cls,
                   const float* __restrict__ gt,
                   const int*   __restrict__ lab,
                   const int*   __restrict__ vidx,
                   float*       __restrict__ blockpart,
                   int V) {
    const int i  = blockIdx.x * BLK + threadIdx.x;
    const int ii = (i < V) ? i : (V - 1);          // clamped: loads stay in-bounds
    const float w = (i < V) ? 1.0f : 0.0f;         // full-EXEC predication

    const int vi = vidx[ii];
    const int lb = lab[vi];

    const float2 lg = *reinterpret_cast<const float2*>(cls  + (size_t)vi * 2);
    const float4 p  = *reinterpret_cast<const float4*>(pred + (size_t)vi * 4);
    const float4 u  = *reinterpret_cast<const float4*>(unc  + (size_t)vi * 4);
    const float4 g  = *reinterpret_cast<const float4*>(gt   + (size_t)vi * 4);

    // ---- classification: -log_softmax picked at label ----
    const float mx  = fmaxf(lg.x, lg.y);
    const float lse = mx + logf(expf(lg.x - mx) + expf(lg.y - mx));
    const float picked = (lb == 1) ? lg.y : lg.x;
    const float a_class = (lse - picked) * w;

    // ---- regression: loss attenuation ----
    const float v0 = fmaf(u.x, u.x, RPN_EPS);
    const float v1 = fmaf(u.y, u.y, RPN_EPS);
    const float v2 = fmaf(u.z, u.z, RPN_EPS);
    const float v3 = fmaf(u.w, u.w, RPN_EPS);
    const float d0 = p.x - g.x;
    const float d1 = p.y - g.y;
    const float d2 = p.z - g.z;
    const float d3 = p.w - g.w;
    const float pos_t = 0.5f * (d0 * d0 / v0 + d1 * d1 / v1 + d2 * d2 / v2 + d3 * d3 / v3)
                      + 0.5f * (logf(v0) + logf(v1) + logf(v2) + logf(v3));
    const float neg_t = 1.0f / v0 + 1.0f / v1 + 1.0f / v2 + 1.0f / v3;

    const float isp = (lb == 1) ? w : 0.0f;
    const float isn = w - isp;

    float r[4];
    if (block_reduce4(a_class, pos_t * isp, neg_t * isn, isp, r)) {
        float* bp = blockpart + (size_t)blockIdx.x * 4;
        bp[0] = r[0]; bp[1] = r[1]; bp[2] = r[2]; bp[3] = r[3];
    }
}

__global__ void __launch_bounds__(BLK)
rpn_final_kernel(const float* __restrict__ blockpart,
                 int nblocks, int V,
                 float* __restrict__ out) {
    float a0 = 0.f, a1 = 0.f, a2 = 0.f, a3 = 0.f;
    for (int b = threadIdx.x; b < nblocks; b += BLK) {   // fixed strided order
        const float* bp = blockpart + (size_t)b * 4;
        a0 += bp[0]; a1 += bp[1]; a2 += bp[2]; a3 += bp[3];
    }
    float r[4];
    if (block_reduce4(a0, a1, a2, a3, r)) {
        const float n_pos = r[3];
        const float n_neg = (float)V - n_pos;
        out[0] = r[0] / (float)V + r[1] / n_pos + r[2] / n_neg;
    }
}

extern "C" void kernel_launch(void* const* d_in, const int* in_sizes, int n_in,
                              void* d_out, int out_size, void* d_ws, size_t ws_size,
                              hipStream_t stream) {
    const float* pred = (const float*)d_in[0];  // bbox_pred            [1,N,4]
    const float* unc  = (const float*)d_in[1];  // bbox_uncertainty_pred[1,N,4]
    const float* cls  = (const float*)d_in[2];  // bbox_class           [1,N,2]
    const float* gt   = (const float*)d_in[3];  // gt_bbox              [1,N,4]
    const int*   lab  = (const int*)d_in[4];    // gt_anchor_label      [1,N]
    const int*   vidx = (const int*)d_in[5];    // valid_indices        [1,V]
    const int V = in_sizes[5];
    const int nblocks = (V + BLK - 1) / BLK;

    float* blockpart = (float*)d_ws;            // nblocks * 4 floats

    rpn_partial_kernel<<<nblocks, BLK, 0, stream>>>(pred, unc, cls, gt, lab, vidx,
                                                    blockpart, V);
    rpn_final_kernel<<<1, BLK, 0, stream>>>(blockpart, nblocks, V, (float*)d_out);
}